// NeRFLoss_51479478010329
// MI455X (gfx1250) — compile-verified
//
#include <hip/hip_runtime.h>
#include <hip/hip_bf16.h>

// MI455X (gfx1250) NeRF loss.
// Memory-bound: 144 MB streamed (fits the 192 MB L2 across graph replays ->
// keep default temporal hints). The distortion loss is restructured so the
// per-sample work is serial FMA inside a lane (issue-bound friendly), and all
// cross-chunk pairwise terms are computed by two v_wmma_f32_16x16x4_f32
// (rank-2 pairwise matrix of the 16 chunk aggregates per ray; ray A in the
// K0/K1 lane-half plane, ray B in K2/K3).

#define LAMBDA_OPACITY    0.001f
#define LAMBDA_DISTORTION 0.001f
#define CHUNK 12                      // samples per lane: 16 lanes * 12 = 192

typedef __attribute__((ext_vector_type(2))) float v2f;
typedef __attribute__((ext_vector_type(8))) float v8f;

// ---------------------------------------------------------------------------
// Kernel 1: per-ray elementwise losses (d_rgb, d_depth, d_opacity).
// ---------------------------------------------------------------------------
__global__ __launch_bounds__(256) void nerf_ray_losses(
    const float* __restrict__ rgb_coarse, const float* __restrict__ rgb_fine,
    const float* __restrict__ rgb_target, const float* __restrict__ depth,
    const float* __restrict__ depth_target, const float* __restrict__ opacity,
    float* __restrict__ out, int n_rays)
{
    int r = blockIdx.x * blockDim.x + threadIdx.x;
    if (r >= n_rays) return;

    float ct0 = rgb_target[3*r+0], ct1 = rgb_target[3*r+1], ct2 = rgb_target[3*r+2];
    float c0 = rgb_coarse[3*r+0] - ct0;
    float c1 = rgb_coarse[3*r+1] - ct1;
    float c2 = rgb_coarse[3*r+2] - ct2;
    float f0 = rgb_fine[3*r+0] - ct0;
    float f1 = rgb_fine[3*r+1] - ct1;
    float f2 = rgb_fine[3*r+2] - ct2;

    out[r] = (c0*c0 + c1*c1 + c2*c2) * (1.0f/3.0f)
           + (f0*f0 + f1*f1 + f2*f2) * (1.0f/3.0f);

    out[n_rays + r] = fabsf(depth[r] - depth_target[r]);

    float o = opacity[r] + 1e-10f;
    out[2*n_rays + r] = LAMBDA_OPACITY * (-o * logf(o));
}

// ---------------------------------------------------------------------------
// Kernel 2: Mip-NeRF 360 distortion loss. One wave = TWO rays.
//   loss_r = 2*sum_{i>j} (wt_i*w_j - w_i*wt_j) + (1/3) sum_i w_i^2 d_i
// Lane l<16 owns 12 consecutive samples of ray A (chunk l); lane l>=16 owns
// chunk l-16 of ray B. Per lane: serial intra-chunk ordered-pair sum S plus
// chunk aggregates (Wc, WTc, wsq). Cross-chunk pairs are the strictly-lower
// triangle of the rank-2 matrix  D[i][j] = WTc_i*Wc_j - Wc_i*WTc_j  over the
// 16 chunks of a ray -> one WMMA f32 16x16x4 per ray:
//   A = {WTc, Wc} (every lane; K0/K1 plane = lanes 0-15 = ray A rows,
//                  K2/K3 plane = lanes 16-31 = ray B rows)
//   B_A = {Wc, -WTc} with hi lanes zeroed  -> D_A = ray-A pair matrix
//   B_B = {Wc, -WTc} with lo lanes zeroed  -> D_B = ray-B pair matrix
// C/D layout: VGPR v -> (M = v + (lane>=16 ? 8:0), N = lane&15).
// ---------------------------------------------------------------------------
__global__ __launch_bounds__(256) void nerf_distortion(
    const float* __restrict__ ws, const float* __restrict__ deltas,
    const float* __restrict__ ts, const int* __restrict__ rays_a,
    float* __restrict__ out, int n_rays)
{
    const int lane   = threadIdx.x & 31;
    const int waveId = (blockIdx.x * blockDim.x + threadIdx.x) >> 5;
    const int rayA   = 2 * waveId;
    if (rayA >= n_rays) return;                // whole-wave uniform guard

    const bool hi    = (lane >= 16);
    const bool hasB  = (rayA + 1 < n_rays);
    const int  myRay = (hi && hasB) ? (rayA + 1) : rayA;

    const int outIdx = rays_a[3 * myRay + 0];
    const int base   = rays_a[3 * myRay + 1];
    const int off    = base + (lane & 15) * CHUNK;   // 16B-aligned (48B/chunk)

    __align__(16) float wbuf[CHUNK];
    __align__(16) float tbuf[CHUNK];
    __align__(16) float dbuf[CHUNK];
    *(float4*)&wbuf[0] = *(const float4*)(ws + off);
    *(float4*)&wbuf[4] = *(const float4*)(ws + off + 4);
    *(float4*)&wbuf[8] = *(const float4*)(ws + off + 8);
    *(float4*)&tbuf[0] = *(const float4*)(ts + off);
    *(float4*)&tbuf[4] = *(const float4*)(ts + off + 4);
    *(float4*)&tbuf[8] = *(const float4*)(ts + off + 8);
    *(float4*)&dbuf[0] = *(const float4*)(deltas + off);
    *(float4*)&dbuf[4] = *(const float4*)(deltas + off + 4);
    *(float4*)&dbuf[8] = *(const float4*)(deltas + off + 8);

    // Serial intra-chunk pass: ordered-pair sum + aggregates (~7 VALU/sample).
    float S = 0.f, Wp = 0.f, WTp = 0.f, wsq = 0.f;
    #pragma unroll
    for (int i = 0; i < CHUNK; ++i) {
        float w  = wbuf[i];
        float wt = w * tbuf[i];
        S   = fmaf(wt, Wp, S);          // + wt_i * sum_{j<i} w_j
        S   = fmaf(-w, WTp, S);         // - w_i  * sum_{j<i} wt_j
        Wp  += w;
        WTp += wt;
        wsq = fmaf(w * w, dbuf[i], wsq);
    }

    // Cross-chunk pairwise terms via two rank-2 WMMAs.
    v2f a;  a.x  = WTp;            a.y  = Wp;     // A[m] = [WTc, Wc]
    v2f bA; bA.x = hi ? 0.f : Wp;  bA.y = hi ? 0.f : -WTp;
    v2f bB; bB.x = hi ? Wp  : 0.f; bB.y = hi ? -WTp : 0.f;
    v8f accA = {0.f,0.f,0.f,0.f,0.f,0.f,0.f,0.f};
    v8f accB = {0.f,0.f,0.f,0.f,0.f,0.f,0.f,0.f};
    accA = __builtin_amdgcn_wmma_f32_16x16x4_f32(
               false, a, false, bA, (short)0, accA, false, false);
    accB = __builtin_amdgcn_wmma_f32_16x16x4_f32(
               false, a, false, bB, (short)0, accB, false, false);

    // Strictly-lower-triangular partial sums of each ray's pair matrix.
    float qA = 0.f, qB = 0.f;
    const int N    = lane & 15;
    const int madd = hi ? 8 : 0;
    #pragma unroll
    for (int v = 0; v < 8; ++v) {
        if (v + madd > N) { qA += accA[v]; qB += accB[v]; }
    }
    qA *= 2.0f;
    qB *= 2.0f;

    // Fold in this lane's serial terms for the ray it belongs to.
    float own = 2.0f * S + wsq * (1.0f/3.0f);
    if (hi) qB += own; else qA += own;

    // Full-wave butterfly reductions (one per ray).
    #pragma unroll
    for (int m = 1; m <= 16; m <<= 1) {
        qA += __shfl_xor(qA, m, 32);
        qB += __shfl_xor(qB, m, 32);
    }

    if (lane == 0)           out[outIdx] = LAMBDA_DISTORTION * qA;
    if (lane == 16 && hasB)  out[outIdx] = LAMBDA_DISTORTION * qB;
}

// ---------------------------------------------------------------------------
extern "C" void kernel_launch(void* const* d_in, const int* in_sizes, int n_in,
                              void* d_out, int out_size, void* d_ws, size_t ws_size,
                              hipStream_t stream) {
    const float* rgb_coarse   = (const float*)d_in[0];
    const float* rgb_fine     = (const float*)d_in[1];
    const float* rgb_target   = (const float*)d_in[2];
    const float* depth        = (const float*)d_in[3];
    const float* depth_target = (const float*)d_in[4];
    const float* opacity      = (const float*)d_in[5];
    const float* ws           = (const float*)d_in[6];
    const float* deltas       = (const float*)d_in[7];
    const float* ts           = (const float*)d_in[8];
    const int*   rays_a       = (const int*)d_in[9];
    float* out = (float*)d_out;

    const int n_rays = in_sizes[3];                    // 65536

    nerf_ray_losses<<<(n_rays + 255) / 256, 256, 0, stream>>>(
        rgb_coarse, rgb_fine, rgb_target, depth, depth_target, opacity,
        out, n_rays);

    const int n_waves = (n_rays + 1) / 2;              // 2 rays per wave32
    const int blocks  = (n_waves + 7) / 8;             // 8 waves per block
    nerf_distortion<<<blocks, 256, 0, stream>>>(
        ws, deltas, ts, rays_a, out + 3 * n_rays, n_rays);
}